// VisibleNet_549755814408
// MI455X (gfx1250) — compile-verified
//
#include <hip/hip_runtime.h>
#include <stdint.h>

// ---------------------------------------------------------------------------
// Problem constants (x: [4,1,128,256,256] f32; out: [4,2,4,256,256] f32)
// ---------------------------------------------------------------------------
#define HW     65536        // 256*256
#define DD     128          // depth
#define NB     4            // batch (c==1)
#define KSEL   4
#define SCALEF 100000.0f

typedef float v4f __attribute__((ext_vector_type(4)));
typedef int   v4i __attribute__((vector_size(4 * sizeof(int))));

// ---------------------------------------------------------------------------
// gfx1250 async-to-LDS path (builtins probe-confirmed; signature from clang
// diagnostic: arg0 = int4 AS1* (global), arg1 = int4 AS3* (LDS), then two
// imm ints (offset, cpol)).
// ---------------------------------------------------------------------------
#ifndef __has_builtin
#define __has_builtin(x) 0
#endif

#if __has_builtin(__builtin_amdgcn_global_load_async_to_lds_b128) && \
    __has_builtin(__builtin_amdgcn_s_wait_asynccnt)
#define USE_ASYNC 1
#else
#define USE_ASYNC 0
#warning "CDNA5_PROBE: async builtins unexpectedly MISSING, using direct-load path"
#endif

#if USE_ASYNC
typedef __attribute__((address_space(1))) v4i glob_v4i;
typedef __attribute__((address_space(3))) v4i lds_v4i;

__device__ __forceinline__ void async_cp_b128(const float* g, const v4f* l) {
  // Generic shared ptr low 32 bits are the LDS byte offset (ISA 10.2);
  // int->pointer casts avoid addrspacecast legality.
  __builtin_amdgcn_global_load_async_to_lds_b128(
      (glob_v4i*)(uintptr_t)g,
      (lds_v4i*)(uint32_t)(uintptr_t)l,
      0, 0);
}
#endif

// ---------------------------------------------------------------------------
// Kernel 0: init per-b {min,max} atomics (uint bit patterns; values >= 0)
// ---------------------------------------------------------------------------
__global__ void init_mm_kernel(unsigned* __restrict__ mm) {
  int t = threadIdx.x;
  if (t < 2 * NB) mm[t] = (t & 1) ? 0u : 0x7f800000u;  // max=0.0f, min=+inf
}

// ---------------------------------------------------------------------------
// Kernel 1: fused relu + top-4 over depth + global min/max atomics.
// One thread owns 4 consecutive w positions (one b128 load per d-slice).
// Writes RAW top-4 values into pred half of out, FINAL depth vals into dep
// half. Grid: (128, NB) x 128 threads (2048 waves).
//
// Deep ASYNCcnt pipeline: 16-slot LDS ring, 15 loads in flight per wave
// (7.5 KB/wave -> ~15 MB in flight device-wide, matching the 23.3 TB/s x
// ~600 ns latency-bandwidth product). All loops rolled (#pragma unroll 1)
// and occupancy target relaxed to avoid VGPR spills.
// ---------------------------------------------------------------------------
#define STAGES 16

__global__ __launch_bounds__(128, 1) void topk_kernel(
    const float* __restrict__ x, float* __restrict__ out,
    unsigned* __restrict__ mm) {
  const int tid = threadIdx.x;
  const int b   = blockIdx.y;
  const int hw4 = (blockIdx.x * 128 + tid) * 4;
  const float* xb = x + (size_t)b * ((size_t)DD * HW) + hw4;

  float tv[4][4];
  int   ti[4][4];
#pragma unroll
  for (int c = 0; c < 4; ++c)
#pragma unroll
    for (int k = 0; k < 4; ++k) { tv[c][k] = -1.0f; ti[c][k] = 0; }
  float vmin = __uint_as_float(0x7f800000u);  // +inf

  auto process = [&](v4f v, int d) {
#pragma unroll
    for (int c = 0; c < 4; ++c) {
      float vv = fmaxf(v[c], 0.0f);           // relu
      vmin = fminf(vmin, vv);
      bool g0 = vv > tv[c][0];
      bool g1 = vv > tv[c][1];
      bool g2 = vv > tv[c][2];
      bool g3 = vv > tv[c][3];
      // stable insert: ties keep the earlier (lower d) index, like lax.top_k
      tv[c][3] = g3 ? (g2 ? tv[c][2] : vv) : tv[c][3];
      ti[c][3] = g3 ? (g2 ? ti[c][2] : d ) : ti[c][3];
      tv[c][2] = g2 ? (g1 ? tv[c][1] : vv) : tv[c][2];
      ti[c][2] = g2 ? (g1 ? ti[c][1] : d ) : ti[c][2];
      tv[c][1] = g1 ? (g0 ? tv[c][0] : vv) : tv[c][1];
      ti[c][1] = g1 ? (g0 ? ti[c][0] : d ) : ti[c][1];
      tv[c][0] = g0 ? vv : tv[c][0];
      ti[c][0] = g0 ? d  : ti[c][0];
    }
  };

#if USE_ASYNC
  // 16-slot LDS ring, 15 loads in flight. Each lane reads back only the 16B
  // it loaded -> no barriers. Issue targets slot (d+15)%16 == the slot read
  // at iteration d-1 (its ds_load already consumed -> no WAR hazard; the
  // memory fences keep the pipeliner from hoisting the issue over the read).
  __shared__ v4f buf[STAGES * 128];
#pragma unroll 1
  for (int s = 0; s < STAGES - 1; ++s)
    async_cp_b128(xb + (size_t)s * HW, &buf[s * 128 + tid]);

#pragma unroll 1
  for (int d = 0; d < DD - STAGES + 1; ++d) {          // d = 0..112
    __builtin_amdgcn_s_wait_asynccnt(STAGES - 2);      // oldest slice landed
    asm volatile("" ::: "memory");
    v4f v = buf[(d & (STAGES - 1)) * 128 + tid];
    process(v, d);
    asm volatile("" ::: "memory");                     // keep issue after read
    async_cp_b128(xb + (size_t)(d + STAGES - 1) * HW,
                  &buf[((d + STAGES - 1) & (STAGES - 1)) * 128 + tid]);
  }
  __builtin_amdgcn_s_wait_asynccnt(0);
  asm volatile("" ::: "memory");
#pragma unroll 1
  for (int d = DD - STAGES + 1; d < DD; ++d) {         // d = 113..127
    v4f v = buf[(d & (STAGES - 1)) * 128 + tid];
    process(v, d);
  }
#else
#pragma unroll 4
  for (int d = 0; d < DD; ++d) {
    v4f v = __builtin_nontemporal_load((const v4f*)(xb + (size_t)d * HW));
    if (d + 8 < DD) __builtin_prefetch(xb + (size_t)(d + 8) * HW, 0, 0);
    process(v, d);
  }
#endif

  // per-b min/max: wave32 reduce, then one atomic per wave
  float vmax = fmaxf(fmaxf(tv[0][0], tv[1][0]), fmaxf(tv[2][0], tv[3][0]));
#pragma unroll
  for (int o = 16; o > 0; o >>= 1) {
    vmin = fminf(vmin, __shfl_down(vmin, o, 32));
    vmax = fmaxf(vmax, __shfl_down(vmax, o, 32));
  }
  if ((tid & 31) == 0) {
    atomicMin(&mm[2 * b],     __float_as_uint(vmin));
    atomicMax(&mm[2 * b + 1], __float_as_uint(vmax));
  }

  // out layout per b: [2][K][HW]; pred = raw vals (normalized later),
  // dep = final (127-idx)/127. Assemble float4 across the 4 owned columns.
  float* ob = out + (size_t)b * (2 * KSEL * (size_t)HW) + hw4;
#pragma unroll
  for (int k = 0; k < KSEL; ++k) {
    v4f p = {tv[0][k], tv[1][k], tv[2][k], tv[3][k]};
    *(v4f*)(ob + (size_t)k * HW) = p;
    v4f dp = {(127.0f - (float)ti[0][k]) / 127.0f,
              (127.0f - (float)ti[1][k]) / 127.0f,
              (127.0f - (float)ti[2][k]) / 127.0f,
              (127.0f - (float)ti[3][k]) / 127.0f};
    *(v4f*)(ob + (size_t)(KSEL + k) * HW) = dp;
  }
}

// ---------------------------------------------------------------------------
// Kernel 2: in-place min-max normalize of the pred half (1M floats).
// ---------------------------------------------------------------------------
__global__ __launch_bounds__(256) void norm_kernel(float* __restrict__ out,
                                                   const unsigned* __restrict__ mm) {
  int t = blockIdx.x * 256 + threadIdx.x;       // 0..262143 float4s
  int b = t >> 16;                              // 65536 float4s per b
  size_t off = (size_t)b * (2 * KSEL * (size_t)HW) + ((size_t)(t & 0xFFFF) * 4);
  float mn  = __uint_as_float(mm[2 * b]);
  float mx  = __uint_as_float(mm[2 * b + 1]);
  float den = (mx - mn) + 1e-15f;
  v4f v = *(const v4f*)(out + off);
  v4f r;
#pragma unroll
  for (int i = 0; i < 4; ++i) r[i] = ((v[i] - mn) / den) * SCALEF;
  *(v4f*)(out + off) = r;
}

// ---------------------------------------------------------------------------
extern "C" void kernel_launch(void* const* d_in, const int* in_sizes, int n_in,
                              void* d_out, int out_size, void* d_ws, size_t ws_size,
                              hipStream_t stream) {
  (void)in_sizes; (void)n_in; (void)out_size; (void)ws_size;
  const float* x  = (const float*)d_in[0];
  float* out      = (float*)d_out;
  unsigned* mm    = (unsigned*)d_ws;            // 8 uints: {min,max} per b

  hipLaunchKernelGGL(init_mm_kernel, dim3(1), dim3(32), 0, stream, mm);
  hipLaunchKernelGGL(topk_kernel, dim3(HW / 4 / 128, NB), dim3(128), 0, stream,
                     x, out, mm);
  hipLaunchKernelGGL(norm_kernel, dim3((NB * HW) / 256), dim3(256), 0, stream,
                     out, mm);
}